// UniMP_FROND_90512140796315
// MI455X (gfx1250) — compile-verified
//
#include <hip/hip_runtime.h>
#include <math.h>

#define N_NODES 50000
#define N_EDGES 1000000
#define DIM 64
#define LN_EPSF 1e-5f

typedef __attribute__((ext_vector_type(2))) float v2f;
typedef __attribute__((ext_vector_type(8))) float v8f;

// Order-preserving int encoding of float for atomicMax-based segment max.
__device__ __forceinline__ int enc_order(float f) {
  int i = __float_as_int(f);
  return (i >= 0) ? i : (i ^ 0x7FFFFFFF);
}
__device__ __forceinline__ float dec_order(int i) {
  return __int_as_float((i >= 0) ? i : (i ^ 0x7FFFFFFF));
}

__global__ void k_init(int* __restrict__ m, float* __restrict__ s, int n) {
  int i = blockIdx.x * blockDim.x + threadIdx.x;
  if (i < n) { m[i] = (int)0x80000000; s[i] = 0.f; }
}

// OUT[rows*64] = H[rows*64] @ W[64*64] + bias  -- fp32 WMMA, one wave per 16x16 tile
__global__ void k_gemm64(const float* __restrict__ H, const float* __restrict__ W,
                         const float* __restrict__ bias, float* __restrict__ OUT,
                         int row_tiles) {
  int wave = (int)((blockIdx.x * (unsigned)blockDim.x + threadIdx.x) >> 5);
  int lane = threadIdx.x & 31;
  int rt   = wave >> 2;          // 4 column tiles (64/16)
  int col0 = (wave & 3) << 4;
  if (rt >= row_tiles) return;   // whole-wave exit; active waves keep EXEC all-ones
  int row0 = rt << 4;
  int m  = lane & 15;
  int kh = lane >> 4;            // 0 or 1: which K-half of the 16x4 A fragment

  v8f acc = {};
  const float* arow = H + (size_t)(row0 + m) * DIM + 2 * kh;
  const float* bcol = W + (size_t)(2 * kh) * DIM + col0 + m;
#pragma unroll
  for (int kb = 0; kb < DIM; kb += 4) {
    v2f a = *(const v2f*)(arow + kb);                // A[m][kb+2kh .. kb+2kh+1]
    v2f b;
    b.x = bcol[(size_t)kb * DIM];                    // W[kb+2kh  ][col0+m]
    b.y = bcol[(size_t)(kb + 1) * DIM];              // W[kb+2kh+1][col0+m]
    acc = __builtin_amdgcn_wmma_f32_16x16x4_f32(
        false, a, false, b, (short)0, acc, false, false);
  }
  float bc = bias[col0 + m];
#pragma unroll
  for (int j = 0; j < 8; ++j)
    OUT[(size_t)(row0 + j + 8 * kh) * DIM + col0 + m] = acc[j] + bc;
}

// One wave per edge: logits[e] = dot(q[dst], k[src]) / 8; atomicMax into m[dst].
__global__ void k_logits(const float* __restrict__ q, const float* __restrict__ k,
                         const int* __restrict__ src, const int* __restrict__ dst,
                         float* __restrict__ logit, int* __restrict__ mmax) {
  int wave = (int)((blockIdx.x * (unsigned)blockDim.x + threadIdx.x) >> 5);
  int lane = threadIdx.x & 31;
  if (wave >= N_EDGES) return;
  int s = src[wave], d = dst[wave];
  v2f qv = *(const v2f*)(q + (size_t)d * DIM + 2 * lane);
  v2f kv = *(const v2f*)(k + (size_t)s * DIM + 2 * lane);
  float p = qv.x * kv.x + qv.y * kv.y;
#pragma unroll
  for (int off = 16; off; off >>= 1) p += __shfl_xor(p, off, 32);
  p *= 0.125f;                       // 1/sqrt(64)
  if (lane == 0) {
    logit[wave] = p;
    atomicMax(mmax + d, enc_order(p));
  }
}

// One thread per edge: e = exp(logit - m[dst]); s[dst] += e; store e in place.
__global__ void k_exp(float* __restrict__ logit, const int* __restrict__ mmax,
                      float* __restrict__ ssum, const int* __restrict__ dst) {
  int e = blockIdx.x * blockDim.x + threadIdx.x;
  if (e >= N_EDGES) return;
  int d = dst[e];
  float ev = expf(logit[e] - dec_order(mmax[d]));
  logit[e] = ev;
  atomicAdd(ssum + d, ev);
}

// One wave per edge: agg[dst] += (e / s[dst]) * v[src]  (2 channels per lane)
__global__ void k_agg(const float* __restrict__ eexp, const float* __restrict__ ssum,
                      const float* __restrict__ v,
                      const int* __restrict__ src, const int* __restrict__ dst,
                      float* __restrict__ agg) {
  int wave = (int)((blockIdx.x * (unsigned)blockDim.x + threadIdx.x) >> 5);
  int lane = threadIdx.x & 31;
  if (wave >= N_EDGES) return;
  int s = src[wave], d = dst[wave];
  float a = eexp[wave] / ssum[d];
  v2f vv = *(const v2f*)(v + (size_t)s * DIM + 2 * lane);
  float* out = agg + (size_t)d * DIM + 2 * lane;
  atomicAdd(out,     a * vv.x);
  atomicAdd(out + 1, a * vv.y);
}

// One wave per node: out = relu(layernorm(agg (+ res)) * g + b)
__global__ void k_ln_relu(const float* __restrict__ agg, const float* __restrict__ res,
                          int use_res, const float* __restrict__ g,
                          const float* __restrict__ b, float* __restrict__ out) {
  int wave = (int)((blockIdx.x * (unsigned)blockDim.x + threadIdx.x) >> 5);
  int lane = threadIdx.x & 31;
  if (wave >= N_NODES) return;
  size_t base = (size_t)wave * DIM + 2 * lane;
  v2f x = *(const v2f*)(agg + base);
  if (use_res) {
    v2f r = *(const v2f*)(res + base);
    x.x += r.x; x.y += r.y;
  }
  float sum = x.x + x.y;
#pragma unroll
  for (int off = 16; off; off >>= 1) sum += __shfl_xor(sum, off, 32);
  float mu = sum * (1.0f / DIM);
  float dx = x.x - mu, dy = x.y - mu;
  float vs = dx * dx + dy * dy;
#pragma unroll
  for (int off = 16; off; off >>= 1) vs += __shfl_xor(vs, off, 32);
  float rs = rsqrtf(vs * (1.0f / DIM) + LN_EPSF);
  float g0 = g[2 * lane], g1 = g[2 * lane + 1];
  float b0 = b[2 * lane], b1 = b[2 * lane + 1];
  v2f o;
  o.x = fmaxf(dx * rs * g0 + b0, 0.f);
  o.y = fmaxf(dy * rs * g1 + b1, 0.f);
  *(v2f*)(out + base) = o;
}

extern "C" void kernel_launch(void* const* d_in, const int* in_sizes, int n_in,
                              void* d_out, int out_size, void* d_ws, size_t ws_size,
                              hipStream_t stream) {
  (void)in_sizes; (void)n_in; (void)out_size; (void)ws_size;
  const float* x    = (const float*)d_in[0];
  const int*   ei   = (const int*)d_in[1];
  const int*   srcp = ei;
  const int*   dstp = ei + N_EDGES;
  const float* Wq0 = (const float*)d_in[2];  const float* bq0 = (const float*)d_in[3];
  const float* Wk0 = (const float*)d_in[4];  const float* bk0 = (const float*)d_in[5];
  const float* Wv0 = (const float*)d_in[6];  const float* bv0 = (const float*)d_in[7];
  const float* Ws0 = (const float*)d_in[8];  const float* bs0 = (const float*)d_in[9];
  const float* g0  = (const float*)d_in[10]; const float* bb0 = (const float*)d_in[11];
  const float* WqL = (const float*)d_in[12]; const float* bqL = (const float*)d_in[13];
  const float* WkL = (const float*)d_in[14]; const float* bkL = (const float*)d_in[15];
  const float* WvL = (const float*)d_in[16]; const float* bvL = (const float*)d_in[17];
  const float* WsL = (const float*)d_in[18]; const float* bsL = (const float*)d_in[19];
  const float* gL  = (const float*)d_in[20]; const float* bbL = (const float*)d_in[21];

  float* ws = (float*)d_ws;
  const size_t ND = (size_t)N_NODES * DIM;
  float* h   = ws;             // current features
  float* q   = h + ND;
  float* k   = q + ND;
  float* v   = k + ND;
  float* agg = v + ND;         // skip + attention aggregate
  float* lg  = agg + ND;       // per-edge logits / exp values (E)
  int*   mm  = (int*)(lg + N_EDGES);     // per-node ordered-int max (N)
  float* ss  = (float*)(mm + N_NODES);   // per-node exp sum (N)

  const int BT = 256;                        // 8 waves per block
  const int row_tiles   = N_NODES / 16;      // 3125 exactly
  const int gemm_blocks = (row_tiles * 4 + 7) / 8;
  const int edge_wb     = (N_EDGES + 7) / 8;
  const int node_wb     = (N_NODES + 7) / 8;
  const int node_tb     = (N_NODES + BT - 1) / BT;
  const int edge_tb     = (N_EDGES + BT - 1) / BT;

  for (int layer = 0; layer < 4; ++layer) {
    const float* hin = (layer == 0) ? x : h;
    const float *Wq_, *bq_, *Wk_, *bk_, *Wv_, *bv_, *Ws_, *bs_, *g_, *b_;
    if (layer == 0) {
      Wq_ = Wq0; bq_ = bq0; Wk_ = Wk0; bk_ = bk0;
      Wv_ = Wv0; bv_ = bv0; Ws_ = Ws0; bs_ = bs0;
      g_ = g0; b_ = bb0;
    } else {
      int i = layer - 1;
      Wq_ = WqL + (size_t)i * DIM * DIM; bq_ = bqL + (size_t)i * DIM;
      Wk_ = WkL + (size_t)i * DIM * DIM; bk_ = bkL + (size_t)i * DIM;
      Wv_ = WvL + (size_t)i * DIM * DIM; bv_ = bvL + (size_t)i * DIM;
      Ws_ = WsL + (size_t)i * DIM * DIM; bs_ = bsL + (size_t)i * DIM;
      g_  = gL  + (size_t)i * DIM;       b_  = bbL + (size_t)i * DIM;
    }

    k_init  <<<node_tb, BT, 0, stream>>>(mm, ss, N_NODES);
    k_gemm64<<<gemm_blocks, BT, 0, stream>>>(hin, Wq_, bq_, q,   row_tiles);
    k_gemm64<<<gemm_blocks, BT, 0, stream>>>(hin, Wk_, bk_, k,   row_tiles);
    k_gemm64<<<gemm_blocks, BT, 0, stream>>>(hin, Wv_, bv_, v,   row_tiles);
    k_gemm64<<<gemm_blocks, BT, 0, stream>>>(hin, Ws_, bs_, agg, row_tiles);
    k_logits<<<edge_wb, BT, 0, stream>>>(q, k, srcp, dstp, lg, mm);
    k_exp   <<<edge_tb, BT, 0, stream>>>(lg, mm, ss, dstp);
    k_agg   <<<edge_wb, BT, 0, stream>>>(lg, ss, v, srcp, dstp, agg);

    float* outp = (layer == 3) ? (float*)d_out : h;
    k_ln_relu<<<node_wb, BT, 0, stream>>>(agg, h, layer > 0 ? 1 : 0, g_, b_, outp);
  }
}